// Softmax_54640573940343
// MI455X (gfx1250) — compile-verified
//
#include <hip/hip_runtime.h>

// Row softmax, x[16384][4096] f32 -> f32, e = exp(x); e / sum(e) (no max-sub).
// Memory-bound: 512 MiB total HBM traffic -> ~22 us floor at 23.3 TB/s.
// gfx1250 paths: GLOBAL_LOAD_ASYNC_TO_LDS_B128 (GVS mode, ASYNCcnt, NT hint)
// for row staging; non-temporal b128 stores for the streaming output.

#define COLS 4096
#define TPB  256
#define PER_THREAD (COLS / TPB)   // 16 floats per thread
#define CHUNKS (PER_THREAD / 4)   // 4 x float4
#define NWAVES (TPB / 32)         // 8 wave32s

typedef float v4f __attribute__((ext_vector_type(4)));

__global__ __launch_bounds__(TPB) void softmax_rows_kernel(
    const float* __restrict__ x, float* __restrict__ out, int rows) {
  // Single shared object -> LDS offset 0; the async-load LDS byte offsets
  // below are relative to the wave's LDS base. Reduction scratch at the tail.
  __shared__ float smem[COLS + NWAVES];

  const int tid = threadIdx.x;
  const int row = blockIdx.x;
  if (row >= rows) return;

  const float* __restrict__ rowp = x + (size_t)row * COLS;
  float* __restrict__ outp = out + (size_t)row * COLS;

  // ---- Stage row into LDS: async tensor-path loads, GVS addressing --------
  // One shared VGPR (tid*16 bytes) is both the LDS dest offset (VDST) and the
  // 32-bit global offset (VADDR); the chunk stride rides the 24-bit IOFFSET,
  // which hardware adds to BOTH addresses. Each instruction moves a contiguous
  // 512B stripe per wave. NT hint: zero-reuse streaming data.
  const unsigned voff = (unsigned)(tid * 16);  // bytes
#define ASYNC_LD(OFF)                                                          \
  asm volatile("global_load_async_to_lds_b128 %0, %1, %2 offset:" #OFF        \
               " th:TH_LOAD_NT"                                               \
               :                                                              \
               : "v"(voff), "v"(voff), "s"(rowp)                              \
               : "memory")
  ASYNC_LD(0);
  ASYNC_LD(4096);
  ASYNC_LD(8192);
  ASYNC_LD(12288);
#undef ASYNC_LD

  // Drain this wave's ASYNCcnt, then barrier so all stripes are visible.
  asm volatile("s_wait_asynccnt 0" ::: "memory");
  __syncthreads();

  // ---- exp() + per-thread partial sum (ds_load_b128 from LDS) -------------
  v4f e[CHUNKS];
  float s = 0.0f;
#pragma unroll
  for (int k = 0; k < CHUNKS; ++k) {
    const int o = k * (TPB * 4) + tid * 4;  // float index in row
    const v4f v = *reinterpret_cast<const v4f*>(&smem[o]);
    e[k].x = __expf(v.x);
    e[k].y = __expf(v.y);
    e[k].z = __expf(v.z);
    e[k].w = __expf(v.w);
    s += (e[k].x + e[k].y) + (e[k].z + e[k].w);
  }

  // ---- wave32 butterfly reduction -----------------------------------------
#pragma unroll
  for (int m = 16; m >= 1; m >>= 1) s += __shfl_xor(s, m, 32);

  // ---- cross-wave reduction via LDS tail scratch --------------------------
  if ((tid & 31) == 0) smem[COLS + (tid >> 5)] = s;
  __syncthreads();
  float tot = 0.0f;
#pragma unroll
  for (int w = 0; w < NWAVES; ++w) tot += smem[COLS + w];

  const float r = __builtin_amdgcn_rcpf(tot);  // v_rcp_f32, ~1 ulp

  // ---- normalize + coalesced non-temporal b128 stores ---------------------
#pragma unroll
  for (int k = 0; k < CHUNKS; ++k) {
    const int o = k * (TPB * 4) + tid * 4;
    v4f d;
    d.x = e[k].x * r;
    d.y = e[k].y * r;
    d.z = e[k].z * r;
    d.w = e[k].w * r;
    __builtin_nontemporal_store(d, reinterpret_cast<v4f*>(outp + o));
  }
}

extern "C" void kernel_launch(void* const* d_in, const int* in_sizes, int n_in,
                              void* d_out, int out_size, void* d_ws, size_t ws_size,
                              hipStream_t stream) {
  (void)n_in; (void)d_ws; (void)ws_size; (void)out_size;
  const float* x = (const float*)d_in[0];
  float* out = (float*)d_out;
  const int rows = in_sizes[0] / COLS;  // 16384 for the reference shape
  softmax_rows_kernel<<<rows, TPB, 0, stream>>>(x, out, rows);
}